// Model_19189913878620
// MI455X (gfx1250) — compile-verified
//
#include <hip/hip_runtime.h>
#include <math.h>
#include <stdint.h>

// ---------------------------------------------------------------------------
// 2-layer bidirectional LSTM LM for MI455X (gfx1250, wave32).
//  - Large GEMMs (input projections + 32000-way head) via V_WMMA_F32_16X16X4_F32
//    (fp32 in / fp32 acc: matches fp32 reference), 32x64 wave tile
//    (2 M-subtiles x 4 N-subtiles) -> 0.75 vmem loads per WMMA.
//  - Loads use the gfx1250 GVS form: uniform SGPR base (advanced by k0,
//    scalar add) + loop-invariant 32-bit per-lane offsets, so the scheduler
//    can clause all 6 loads per K-step instead of serializing on 64-bit
//    divergent address chains (round-2/3 asm pathology).
//  - Head C-stores are nontemporal (TH=NT) so the 524 MB output stream does
//    not evict Wlin (131 MB, L2-resident) from the 192 MB L2.
//  - LSTM recurrence: 1024-thread workgroup per direction, h in LDS,
//    dot products via float2 + __builtin_elementwise_fma -> v_pk_fma_f32.
// ---------------------------------------------------------------------------

typedef __attribute__((ext_vector_type(2))) float v2f;
typedef __attribute__((ext_vector_type(8))) float v8f;

#define SEQ    4096
#define EMBED  512
#define HIDDEN 512
#define VOCAB  32000

// ---------------------------------------------------------------------------
// Embedding gather: x[s, 0:512] = embed_w[input[s], 0:512]
// ---------------------------------------------------------------------------
__global__ __launch_bounds__(256) void embed_kernel(const int* __restrict__ idx,
                                                    const float* __restrict__ emb,
                                                    float* __restrict__ x) {
  const int s = blockIdx.x;
  const int r = idx[s];
  const float2* src = (const float2*)(emb + (size_t)r * EMBED);
  float2* dst = (float2*)(x + (size_t)s * EMBED);
  dst[threadIdx.x] = src[threadIdx.x];  // 256 threads * float2 = 512 floats
}

// ---------------------------------------------------------------------------
// C[M,N] = A[M,K] @ B[N,K]^T + bias[N]      (A row-major, B = weight [N,K])
// grid = (N/64, M/256), block = 256 threads = 8 waves.
// Wave computes a 32(M) x 64(N) tile: 2x4 WMMA accumulators; each A fragment
// reused over 4 N-subtiles, each B fragment reused over 2 M-subtiles.
//
// V_WMMA_F32_16X16X4_F32 fragment layouts (ISA 7.12.2):
//   A 16x4 : lane<16 -> M=lane,   K={k0,k0+1};  lane>=16 -> M=lane-16, K={k0+2,k0+3}
//   B 4x16 : lane<16 -> N=lane,   K={k0,k0+1};  lane>=16 -> N=lane-16, K={k0+2,k0+3}
//   C/D    : vgpr v -> row v (lanes 0-15) / row v+8 (lanes 16-31), col = lane&15
// ---------------------------------------------------------------------------
template <bool NT>
__global__ __launch_bounds__(256) void gemm_wmma_bias(
    const float* __restrict__ A, const float* __restrict__ B,
    const float* __restrict__ bias, float* __restrict__ C,
    int K, int ldc) {
  const int lane = threadIdx.x & 31;
  const int wv   = threadIdx.x >> 5;
  const int half = lane >> 4;   // 0: lanes 0-15, 1: lanes 16-31
  const int col  = lane & 15;
  const int m0   = blockIdx.y * 256 + wv * 32;
  const int n0   = blockIdx.x * 64;

  // Loop-invariant 32-bit per-lane offsets (elements); all matrices < 2^31
  // elements, so row*K fits u32.  Combined with the uniform (A + k0)/(B + k0)
  // bases below this selects VMEM GVS addressing: SGPR base + VGPR u32 offset.
  const uint32_t uK  = (uint32_t)K;
  const uint32_t oa0 = (uint32_t)(m0 + col) * uK + 2u * (uint32_t)half;
  const uint32_t oa1 = oa0 + 16u * uK;
  const uint32_t ob0 = (uint32_t)(n0 + col) * uK + 2u * (uint32_t)half;
  const uint32_t ob1 = ob0 + 16u * uK;
  const uint32_t ob2 = ob0 + 32u * uK;
  const uint32_t ob3 = ob0 + 48u * uK;

  v8f acc00 = {}, acc01 = {}, acc02 = {}, acc03 = {};
  v8f acc10 = {}, acc11 = {}, acc12 = {}, acc13 = {};

#pragma unroll 2
  for (int k0 = 0; k0 < K; k0 += 4) {
    const float* Ak = A + k0;   // uniform -> SGPR base, scalar increment
    const float* Bk = B + k0;   // uniform -> SGPR base, scalar increment
    const v2f a0 = *(const v2f*)(Ak + oa0);   // 8B-aligned (K % 4 == 0)
    const v2f a1 = *(const v2f*)(Ak + oa1);
    const v2f b0 = *(const v2f*)(Bk + ob0);
    const v2f b1 = *(const v2f*)(Bk + ob1);
    const v2f b2 = *(const v2f*)(Bk + ob2);
    const v2f b3 = *(const v2f*)(Bk + ob3);
    acc00 = __builtin_amdgcn_wmma_f32_16x16x4_f32(false, a0, false, b0, (short)0, acc00, false, false);
    acc10 = __builtin_amdgcn_wmma_f32_16x16x4_f32(false, a1, false, b0, (short)0, acc10, false, false);
    acc01 = __builtin_amdgcn_wmma_f32_16x16x4_f32(false, a0, false, b1, (short)0, acc01, false, false);
    acc11 = __builtin_amdgcn_wmma_f32_16x16x4_f32(false, a1, false, b1, (short)0, acc11, false, false);
    acc02 = __builtin_amdgcn_wmma_f32_16x16x4_f32(false, a0, false, b2, (short)0, acc02, false, false);
    acc12 = __builtin_amdgcn_wmma_f32_16x16x4_f32(false, a1, false, b2, (short)0, acc12, false, false);
    acc03 = __builtin_amdgcn_wmma_f32_16x16x4_f32(false, a0, false, b3, (short)0, acc03, false, false);
    acc13 = __builtin_amdgcn_wmma_f32_16x16x4_f32(false, a1, false, b3, (short)0, acc13, false, false);
  }

  const float bb0 = bias[n0 + col];
  const float bb1 = bias[n0 + 16 + col];
  const float bb2 = bias[n0 + 32 + col];
  const float bb3 = bias[n0 + 48 + col];
  const int mrow = m0 + 8 * half;

#pragma unroll
  for (int v = 0; v < 8; ++v) {
    const size_t base0 = (size_t)(mrow + v) * (size_t)ldc + n0;       // M-subtile 0
    const size_t base1 = (size_t)(mrow + 16 + v) * (size_t)ldc + n0;  // M-subtile 1
    const float o00 = acc00[v] + bb0, o01 = acc01[v] + bb1;
    const float o02 = acc02[v] + bb2, o03 = acc03[v] + bb3;
    const float o10 = acc10[v] + bb0, o11 = acc11[v] + bb1;
    const float o12 = acc12[v] + bb2, o13 = acc13[v] + bb3;
    if (NT) {
      __builtin_nontemporal_store(o00, &C[base0 + col]);
      __builtin_nontemporal_store(o01, &C[base0 + 16 + col]);
      __builtin_nontemporal_store(o02, &C[base0 + 32 + col]);
      __builtin_nontemporal_store(o03, &C[base0 + 48 + col]);
      __builtin_nontemporal_store(o10, &C[base1 + col]);
      __builtin_nontemporal_store(o11, &C[base1 + 16 + col]);
      __builtin_nontemporal_store(o12, &C[base1 + 32 + col]);
      __builtin_nontemporal_store(o13, &C[base1 + 48 + col]);
    } else {
      C[base0 + col]      = o00;
      C[base0 + 16 + col] = o01;
      C[base0 + 32 + col] = o02;
      C[base0 + 48 + col] = o03;
      C[base1 + col]      = o10;
      C[base1 + 16 + col] = o11;
      C[base1 + 32 + col] = o12;
      C[base1 + 48 + col] = o13;
    }
  }
}

// ---------------------------------------------------------------------------
// LSTM scan, one direction per block (blockIdx.x: 0=fwd, 1=bwd), 1024 threads.
// xproj = x @ Wih^T + b is precomputed ([SEQ, 4H], gate order i,f,g,o).
// Thread tid owns gate rows r0=tid and r1=tid+1024:
//   tid <  512 : i[tid],     g[tid]
//   tid >= 512 : f[tid-512], o[tid-512]   (shipped to owner via LDS)
// Dot products use float2 fma -> v_pk_fma_f32 (2 fp32 MACs/lane/instr).
// ---------------------------------------------------------------------------
__device__ __forceinline__ float sigm(float x) { return 1.0f / (1.0f + __expf(-x)); }

__global__ __launch_bounds__(1024) void lstm_scan(
    const float* __restrict__ xpf, const float* __restrict__ xpb,
    const float* __restrict__ Whhf, const float* __restrict__ Whhb,
    float* __restrict__ out /* [SEQ, 2H] */) {
  const int dir = blockIdx.x;
  const float* __restrict__ xp = dir ? xpb : xpf;
  const float* __restrict__ W  = dir ? Whhb : Whhf;
  const int tid = threadIdx.x;

  __shared__ float h_sh[HIDDEN];
  __shared__ float fo_sh[2 * HIDDEN];  // [0:H)=f, [H:2H)=o

  if (tid < HIDDEN) h_sh[tid] = 0.0f;
  float c = 0.0f;
  __syncthreads();

  const float* __restrict__ w0 = W + (size_t)tid * HIDDEN;
  const float* __restrict__ w1 = W + (size_t)(tid + 1024) * HIDDEN;

  for (int step = 0; step < SEQ; ++step) {
    const int t = dir ? (SEQ - 1 - step) : step;
    v2f a0 = { xp[(size_t)t * 2048 + tid], 0.0f };
    v2f a1 = { xp[(size_t)t * 2048 + tid + 1024], 0.0f };

#pragma unroll 4
    for (int e = 0; e < HIDDEN; e += 4) {
      const v2f* hp  = (const v2f*)(h_sh + e);  // LDS broadcast (uniform addr)
      const v2f* w0p = (const v2f*)(w0 + e);
      const v2f* w1p = (const v2f*)(w1 + e);
      const v2f h01 = hp[0];
      const v2f h23 = hp[1];
      a0 = __builtin_elementwise_fma(h01, w0p[0], a0);  // v_pk_fma_f32
      a0 = __builtin_elementwise_fma(h23, w0p[1], a0);
      a1 = __builtin_elementwise_fma(h01, w1p[0], a1);
      a1 = __builtin_elementwise_fma(h23, w1p[1], a1);
    }
    const float acc0 = a0.x + a0.y;
    const float acc1 = a1.x + a1.y;

    __syncthreads();                       // all h_sh reads done
    if (tid >= HIDDEN) {
      const int j = tid - HIDDEN;
      fo_sh[j]          = sigm(acc0);      // f
      fo_sh[HIDDEN + j] = sigm(acc1);      // o
    }
    __syncthreads();
    if (tid < HIDDEN) {
      const float ig = sigm(acc0);
      const float gg = tanhf(acc1);
      const float fg = fo_sh[tid];
      const float og = fo_sh[HIDDEN + tid];
      c = fg * c + ig * gg;
      const float hv = og * tanhf(c);
      h_sh[tid] = hv;
      out[(size_t)t * (2 * HIDDEN) + (size_t)dir * HIDDEN + tid] = hv;
    }
    __syncthreads();                       // h_sh update visible next step
  }
}

// ---------------------------------------------------------------------------
extern "C" void kernel_launch(void* const* d_in, const int* in_sizes, int n_in,
                              void* d_out, int out_size, void* d_ws, size_t ws_size,
                              hipStream_t stream) {
  (void)in_sizes; (void)n_in; (void)out_size; (void)ws_size;
  const int*   input = (const int*)d_in[0];
  const float* embw  = (const float*)d_in[1];
  const float* Wih0f = (const float*)d_in[2];
  const float* Whh0f = (const float*)d_in[3];
  const float* b0f   = (const float*)d_in[4];
  const float* Wih0b = (const float*)d_in[5];
  const float* Whh0b = (const float*)d_in[6];
  const float* b0b   = (const float*)d_in[7];
  const float* Wih1f = (const float*)d_in[8];
  const float* Whh1f = (const float*)d_in[9];
  const float* b1f   = (const float*)d_in[10];
  const float* Wih1b = (const float*)d_in[11];
  const float* Whh1b = (const float*)d_in[12];
  const float* b1b   = (const float*)d_in[13];
  const float* Wlin  = (const float*)d_in[14];
  const float* blin  = (const float*)d_in[15];
  float* out = (float*)d_out;

  // Workspace carve-up (floats): all regions fully written before first read.
  float* ws  = (float*)d_ws;
  float* x0  = ws;                            // [SEQ, EMBED]
  float* xpf = x0  + (size_t)SEQ * EMBED;     // [SEQ, 4H] fwd (reused L0/L1)
  float* xpb = xpf + (size_t)SEQ * 2048;      // [SEQ, 4H] bwd (reused L0/L1)
  float* y0  = xpb + (size_t)SEQ * 2048;      // [SEQ, 2H]
  float* y1  = y0  + (size_t)SEQ * 1024;      // [SEQ, 2H]

  const dim3 blk(256);
  const dim3 gridProj(2048 / 64, SEQ / 256);  // N=2048 tiles x M tiles (32x16)
  const dim3 gridHead(VOCAB / 64, SEQ / 256); // N=32000 tiles x M tiles (500x16)

  embed_kernel<<<SEQ, 256, 0, stream>>>(input, embw, x0);

  // layer 0: input projections (K=512), then bidirectional scan
  gemm_wmma_bias<false><<<gridProj, blk, 0, stream>>>(x0, Wih0f, b0f, xpf, EMBED, 2048);
  gemm_wmma_bias<false><<<gridProj, blk, 0, stream>>>(x0, Wih0b, b0b, xpb, EMBED, 2048);
  lstm_scan<<<2, 1024, 0, stream>>>(xpf, xpb, Whh0f, Whh0b, y0);

  // layer 1: input projections (K=1024), then bidirectional scan
  gemm_wmma_bias<false><<<gridProj, blk, 0, stream>>>(y0, Wih1f, b1f, xpf, 1024, 2048);
  gemm_wmma_bias<false><<<gridProj, blk, 0, stream>>>(y0, Wih1b, b1b, xpb, 1024, 2048);
  lstm_scan<<<2, 1024, 0, stream>>>(xpf, xpb, Whh1f, Whh1b, y1);

  // output head: [4096,1024] @ [1024,32000]^T + blin (268 GFLOP, WMMA-bound);
  // nontemporal stores keep Wlin resident in L2 across the whole GEMM.
  gemm_wmma_bias<true><<<gridHead, blk, 0, stream>>>(y1, Wlin, blin, out, 1024, VOCAB);
}